// LSTM_CELL_69123203661994
// MI455X (gfx1250) — compile-verified
//
#include <hip/hip_runtime.h>
#include <hip/hip_bf16.h>

// LSTM-cell with per-gate MLPs on MI455X (gfx1250).
// Pure bandwidth problem: W1 (157MB) + W3 (52MB) streamed once ~= 9us at 23.3TB/s.
// GEMV for W1 is done with V_WMMA_F32_16X16X4_F32: A = 16 rows x 4 K of W1,
// B = input-vector chunk broadcast across N columns -> every D column holds the
// row dot-product. Inner loop = 2x global_load_b64 + 1 wmma per 256B of W1.

typedef float v2f __attribute__((ext_vector_type(2)));
typedef float v8f __attribute__((ext_vector_type(8)));

#define STATE   65536
#define INPUT   32768
#define CONCAT  (STATE + INPUT)   // 98304
#define H1      100
#define H2      50
#define NG      4
#define ROWS1   (NG * H1)         // 400 rows of W1 (flattened [g,h])
#define TILES1  (ROWS1 / 16)      // 25 exact
#define KCHUNKS 96
#define KPER    (CONCAT / KCHUNKS) // 1024
#define WAVES_A 4
#define KPW     (KPER / WAVES_A)   // 256 K per wave -> 64 wmma iters

// ---------------------------------------------------------------------------
// Kernel A: partial GEMV  h1_partial[row][chunk] = sum_{k in chunk} W1[row][k]*c[k]
// c = [hyp ; x]: a wave's 256-K range never straddles the 65536 boundary
// (k0 % 256 == 0, 65536 % 256 == 0), so the side-select is hoisted out of
// the loop entirely -> inner loop is pure  load_b64 x2 + wmma.
// f32 WMMA operand layout (ISA 7.12.2, 32-bit A 16x4):
//   lane L<16 : A vgpr0 = A[L][0],  vgpr1 = A[L][1]
//   lane L>=16: A vgpr0 = A[L-16][2], vgpr1 = A[L-16][3]
// -> each lane: one float2 load at row (lane&15), col k + 2*(lane>>4).
// B (4x16) mirrors this in K, broadcast over N -> same float2 load from c.
// D (16x16 f32): vgpr r, lane 0 = rowsum[r], lane 16 = rowsum[r+8].
// ---------------------------------------------------------------------------
__global__ __launch_bounds__(WAVES_A * 32) void k_h1_wmma(
    const float* __restrict__ W1,
    const float* __restrict__ hyp,
    const float* __restrict__ x,
    float* __restrict__ partials)
{
    const int lane = threadIdx.x & 31;
    const int wave = threadIdx.x >> 5;
    const int tile = blockIdx.x;        // 0..24
    const int chunk = blockIdx.y;       // 0..95
    const int m  = lane & 15;           // row within tile
    const int hi = lane >> 4;           // 0: K0/K1 half, 1: K2/K3 half
    const int row = tile * 16 + m;
    const int k0 = chunk * KPER + wave * KPW;

    // Hoisted base pointers: everything in the loop is base + 4*i.
    const float* aP = W1 + (long)row * CONCAT + k0 + 2 * hi;
    const float* bP = ((k0 < STATE) ? (hyp + k0) : (x + (k0 - STATE))) + 2 * hi;

    v8f acc0 = {}; v8f acc1 = {};

#pragma unroll 8
    for (int i = 0; i < KPW / 4; ++i) {
        v2f a = *(const v2f*)(aP + 4 * i);   // W1[row][k .. k+1] (+2 for hi lanes)
        v2f b = *(const v2f*)(bP + 4 * i);   // c[k .. k+1]       (+2 for hi lanes)
        if (i & 1)
            acc1 = __builtin_amdgcn_wmma_f32_16x16x4_f32(false, a, false, b,
                                                         (short)0, acc1, false, false);
        else
            acc0 = __builtin_amdgcn_wmma_f32_16x16x4_f32(false, a, false, b,
                                                         (short)0, acc0, false, false);
    }

    // lanes 0 and 16 hold the 16 row sums (columns are replicated)
    __shared__ float red[WAVES_A][16];
    if (m == 0) {
#pragma unroll
        for (int r = 0; r < 8; ++r)
            red[wave][hi * 8 + r] = acc0[r] + acc1[r];
    }
    __syncthreads();

    const int t = threadIdx.x;
    if (t < 16) {
        float p = red[0][t] + red[1][t] + red[2][t] + red[3][t];
        partials[(long)(tile * 16 + t) * KCHUNKS + chunk] = p;
    }
}

// ---------------------------------------------------------------------------
// Kernel A2: h1[row] = relu(sum_chunks partials + b1)   (deterministic sequential sum)
// ---------------------------------------------------------------------------
__global__ __launch_bounds__(128) void k_h1_finish(
    const float* __restrict__ partials,
    const float* __restrict__ b1,
    float* __restrict__ h1)
{
    const int row = blockIdx.x * 128 + threadIdx.x;
    if (row >= ROWS1) return;
    float s = b1[row];
#pragma unroll 8
    for (int c = 0; c < KCHUNKS; ++c) s += partials[(long)row * KCHUNKS + c];
    h1[row] = s > 0.f ? s : 0.f;
}

// ---------------------------------------------------------------------------
// Kernel B: h2 = relu(W2 @ h1 + b2), 200 outputs x dot-100. Tiny.
// ---------------------------------------------------------------------------
__global__ __launch_bounds__(256) void k_h2(
    const float* __restrict__ W2,
    const float* __restrict__ b2,
    const float* __restrict__ h1,
    float* __restrict__ h2)
{
    __shared__ float sh1[ROWS1];
    const int t = threadIdx.x;
    for (int i = t; i < ROWS1; i += 256) sh1[i] = h1[i];
    __syncthreads();
    if (t < NG * H2) {
        const int g = t / H2;
        float acc = b2[t];                        // b2 flat [g*50+j] == t
        const float* wr = W2 + (long)t * H1;      // (g*50+j)*100
#pragma unroll 4
        for (int k = 0; k < H1; ++k) acc += wr[k] * sh1[g * H1 + k];
        h2[t] = acc > 0.f ? acc : 0.f;
    }
}

// ---------------------------------------------------------------------------
// Kernel C: z[g][s] = W3[g][s][:]·h2[g] + b3[g][s], then gate math.
// Each thread reads 4 contiguous 200B rows (full cacheline utilization).
// ---------------------------------------------------------------------------
__device__ __forceinline__ float sigmoidf_(float v) { return 1.f / (1.f + __expf(-v)); }

__global__ __launch_bounds__(256) void k_gates(
    const float* __restrict__ W3,
    const float* __restrict__ b3,
    const float* __restrict__ h2,
    const float* __restrict__ cprev,
    float* __restrict__ out)
{
    __shared__ float sh2[NG * H2];
    const int t = threadIdx.x;
    if (t < NG * H2) sh2[t] = h2[t];
    __syncthreads();

    const int s = blockIdx.x * 256 + t;
    float z[NG];
#pragma unroll
    for (int g = 0; g < NG; ++g) {
        const v2f* wr = (const v2f*)(W3 + ((long)g * STATE + s) * (long)H2); // 200B rows, 8B aligned
        float acc = b3[(long)g * STATE + s];
#pragma unroll
        for (int k = 0; k < H2 / 2; ++k) {
            v2f w = wr[k];
            acc += w.x * sh2[g * H2 + 2 * k] + w.y * sh2[g * H2 + 2 * k + 1];
        }
        z[g] = acc;
    }
    const float fg = sigmoidf_(z[0]);
    const float ig = sigmoidf_(z[1]);
    const float ch = tanhf(z[2]);
    const float og = sigmoidf_(z[3]);
    const float nc = fg * cprev[s] + ig * ch;
    const float nh = tanhf(nc) * og;
    out[s] = nh;              // new_hypothesis (tuple element 0)
    out[STATE + s] = nc;      // new_cell_state (tuple element 1)
}

// ---------------------------------------------------------------------------
extern "C" void kernel_launch(void* const* d_in, const int* in_sizes, int n_in,
                              void* d_out, int out_size, void* d_ws, size_t ws_size,
                              hipStream_t stream) {
    const float* cprev = (const float*)d_in[0];
    const float* hyp   = (const float*)d_in[1];
    const float* x     = (const float*)d_in[2];
    const float* W1    = (const float*)d_in[3];
    const float* b1    = (const float*)d_in[4];
    const float* W2    = (const float*)d_in[5];
    const float* b2    = (const float*)d_in[6];
    const float* W3    = (const float*)d_in[7];
    const float* b3    = (const float*)d_in[8];
    float* out = (float*)d_out;

    float* ws       = (float*)d_ws;
    float* partials = ws;                          // 400 * 96 floats
    float* h1       = partials + ROWS1 * KCHUNKS;  // 400 floats
    float* h2       = h1 + ROWS1;                  // 200 floats

    k_h1_wmma  <<<dim3(TILES1, KCHUNKS), WAVES_A * 32, 0, stream>>>(W1, hyp, x, partials);
    k_h1_finish<<<(ROWS1 + 127) / 128, 128, 0, stream>>>(partials, b1, h1);
    k_h2       <<<1, 256, 0, stream>>>(W2, b2, h1, h2);
    k_gates    <<<STATE / 256, 256, 0, stream>>>(W3, b3, h2, cprev, out);
}